// LocalFeatureAggregation_66958540144768
// MI455X (gfx1250) — compile-verified
//
#include <hip/hip_runtime.h>
#include <cstddef>

typedef __attribute__((ext_vector_type(2))) float v2f;
typedef __attribute__((ext_vector_type(8))) float v8f;

#define NEG_SLOPE 0.2f
#define BN_EPS    1e-5f
#define TS        68          // LDS tile row stride in floats (bank-conflict-free b64 A-frags)
#define NW        8           // waves per 256-thread block (wave32)
#define B_DIM     2
#define N_PTS     65536
#define NTILES_BIG   (B_DIM * N_PTS)      // 131072 tiles of 16 rows over B*N*K rows
#define NROWS_BIG    (NTILES_BIG * 16)    // 2,097,152
#define NROWS_SMALL  (B_DIM * N_PTS)      // 131072 rows for pooled / cached heads
#define NTILES_SMALL (NROWS_SMALL / 16)   // 8192

__device__ __forceinline__ v8f vzero8() {
    v8f z = {0.f,0.f,0.f,0.f,0.f,0.f,0.f,0.f};
    return z;
}
__device__ __forceinline__ float lrelu(float x) { return x > 0.f ? x : NEG_SLOPE * x; }

// D = A(16x4) * B(4x16) + C, fp32, wave32
__device__ __forceinline__ v8f wmma4(v2f a, v2f b, v8f c) {
    return __builtin_amdgcn_wmma_f32_16x16x4_f32(false, a, false, b, (short)0, c, false, false);
}

// Weights live in LDS packed in K-pairs: sW[((k>>1)*C + c)*2 + (k&1)]
// so a B-fragment (W[k,c], W[k+1,c]) is one aligned 8-byte LDS load.
__device__ __forceinline__ v2f bfrag(const float* sW, int kk /*=k>>1*/, int c, int C) {
    return *(const v2f*)&sW[(kk * C + c) * 2];
}

// ---------------------------------------------------------------------------
__global__ void zero_ws_kernel(float* __restrict__ w, int n) {
    int i = blockIdx.x * blockDim.x + threadIdx.x;
    if (i < n) w[i] = 0.f;
}

// acc[0..C)=sum, acc[64..64+C)=sumsq  ->  scale/shift
__global__ void finalize_stats_kernel(const float* __restrict__ acc,
                                      const float* __restrict__ g,
                                      const float* __restrict__ be,
                                      float* __restrict__ sc,
                                      float* __restrict__ sh,
                                      float invM, int C) {
    int c = threadIdx.x;
    if (c < C) {
        float mean = acc[c] * invM;
        float var  = acc[64 + c] * invM - mean * mean;
        float s    = g[c] * rsqrtf(var + BN_EPS);
        sc[c] = s;
        sh[c] = be[c] - mean * s;
    }
}

// ---------------------------------------------------------------------------
// Fused chain over [B,N,K] rows in 16-row tiles (one tile == one (b,n)).
// STAGE==1: stats of y1 = raw@W1+b1
// STAGE==2: stats of y2 = h1@W2+b2
// STAGE==3: stats of y3 = [gather|h2]@Wn+bn
// STAGE==4: feat3 -> attn softmax over K -> pooled [B*N,64]
// ---------------------------------------------------------------------------
template<int STAGE>
__global__ __launch_bounds__(256) void stage_kernel(
    const float* __restrict__ raw, const float* __restrict__ cached,
    const int* __restrict__ nidx,
    const float* __restrict__ W1, const float* __restrict__ b1,
    const float* __restrict__ W2, const float* __restrict__ b2,
    const float* __restrict__ Wn, const float* __restrict__ bn,
    const float* __restrict__ Wat,
    const float* __restrict__ ss, float* __restrict__ acc,
    float* __restrict__ pooled)
{
    __shared__ float sW1[12 * 32];
    __shared__ float sW2[32 * 32];
    __shared__ float sWn[64 * 64];
    __shared__ float sWat[64 * 64];
    __shared__ float sB1[32], sB2[32], sBn[64];
    __shared__ float sSc1[32], sSh1[32], sSc2[32], sSh2[32], sSc3[64], sSh3[64];
    __shared__ float sT[NW][16 * TS];
    __shared__ float sSum[64], sSsq[64];

    const int t = threadIdx.x;
    // pack weights in K-pairs
    for (int i = t; i < 12 * 32; i += 256) {
        const int k = i >> 5, c = i & 31;
        sW1[(((k >> 1) * 32) + c) * 2 + (k & 1)] = (i < 320) ? W1[i] : 0.f;
    }
    if (STAGE >= 2) for (int i = t; i < 32 * 32; i += 256) {
        const int k = i >> 5, c = i & 31;
        sW2[(((k >> 1) * 32) + c) * 2 + (k & 1)] = W2[i];
    }
    if (STAGE >= 3) for (int i = t; i < 64 * 64; i += 256) {
        const int k = i >> 6, c = i & 63;
        sWn[(((k >> 1) * 64) + c) * 2 + (k & 1)] = Wn[i];
    }
    if (STAGE == 4) for (int i = t; i < 64 * 64; i += 256) {
        const int k = i >> 6, c = i & 63;
        sWat[(((k >> 1) * 64) + c) * 2 + (k & 1)] = Wat[i];
    }
    if (t < 32) {
        sB1[t] = b1[t];
        if (STAGE >= 2) { sB2[t] = b2[t]; sSc1[t] = ss[t]; sSh1[t] = ss[64 + t]; }
        if (STAGE >= 3) { sSc2[t] = ss[128 + t]; sSh2[t] = ss[192 + t]; }
    }
    if (t < 64) {
        if (STAGE >= 3) sBn[t] = bn[t];
        if (STAGE == 4) { sSc3[t] = ss[256 + t]; sSh3[t] = ss[320 + t]; }
        if (STAGE < 4)  { sSum[t] = 0.f; sSsq[t] = 0.f; }
    }
    __syncthreads();

    const int lane = t & 31, wid = t >> 5;
    const int col = lane & 15, hi = lane >> 4;
    float* myT = sT[wid];
    float ac_s[4] = {0, 0, 0, 0}, ac_q[4] = {0, 0, 0, 0};

    for (int tile = blockIdx.x * NW + wid; tile < NTILES_BIG; tile += gridDim.x * NW) {
        // ---- GEMM1: raw[16x10, pad->12] @ W1[10x32] ----
        v8f c1[2]; c1[0] = vzero8(); c1[1] = vzero8();
        const float* rawRow = raw + (size_t)(tile * 16 + col) * 10;
        #pragma unroll
        for (int kst = 0; kst < 3; ++kst) {
            const int k = kst * 4 + 2 * hi;
            v2f a;
            a.x = (k < 10)     ? rawRow[k]     : 0.f;
            a.y = (k + 1 < 10) ? rawRow[k + 1] : 0.f;
            #pragma unroll
            for (int ct = 0; ct < 2; ++ct)
                c1[ct] = wmma4(a, bfrag(sW1, kst * 2 + hi, ct * 16 + col, 32), c1[ct]);
        }
        #pragma unroll
        for (int ct = 0; ct < 2; ++ct) {
            const float bb = sB1[ct * 16 + col];
            #pragma unroll
            for (int v = 0; v < 8; ++v) c1[ct][v] += bb;
        }
        if (STAGE == 1) {
            #pragma unroll
            for (int ct = 0; ct < 2; ++ct) {
                float s = 0.f, q = 0.f;
                #pragma unroll
                for (int v = 0; v < 8; ++v) { float x = c1[ct][v]; s += x; q += x * x; }
                ac_s[ct] += s; ac_q[ct] += q;
            }
            continue;
        }
        // h1 = lrelu(affine1(y1)) -> LDS [16x32]
        #pragma unroll
        for (int ct = 0; ct < 2; ++ct) {
            const float sc = sSc1[ct * 16 + col], sh = sSh1[ct * 16 + col];
            #pragma unroll
            for (int v = 0; v < 8; ++v)
                myT[(v + 8 * hi) * TS + ct * 16 + col] = lrelu(c1[ct][v] * sc + sh);
        }
        // ---- GEMM2: h1[16x32] @ W2[32x32] ----
        v8f c2[2]; c2[0] = vzero8(); c2[1] = vzero8();
        #pragma unroll
        for (int kst = 0; kst < 8; ++kst) {
            const v2f a = *(const v2f*)&myT[col * TS + kst * 4 + 2 * hi];
            #pragma unroll
            for (int ct = 0; ct < 2; ++ct)
                c2[ct] = wmma4(a, bfrag(sW2, kst * 2 + hi, ct * 16 + col, 32), c2[ct]);
        }
        #pragma unroll
        for (int ct = 0; ct < 2; ++ct) {
            const float bb = sB2[ct * 16 + col];
            #pragma unroll
            for (int v = 0; v < 8; ++v) c2[ct][v] += bb;
        }
        if (STAGE == 2) {
            #pragma unroll
            for (int ct = 0; ct < 2; ++ct) {
                float s = 0.f, q = 0.f;
                #pragma unroll
                for (int v = 0; v < 8; ++v) { float x = c2[ct][v]; s += x; q += x * x; }
                ac_s[ct] += s; ac_q[ct] += q;
            }
            continue;
        }
        // h2 -> feat cols 32..63
        #pragma unroll
        for (int ct = 0; ct < 2; ++ct) {
            const float sc = sSc2[ct * 16 + col], sh = sSh2[ct * 16 + col];
            #pragma unroll
            for (int v = 0; v < 8; ++v)
                myT[(v + 8 * hi) * TS + 32 + ct * 16 + col] = lrelu(c2[ct][v] * sc + sh);
        }
        // gather neighbor cached features -> feat cols 0..31 (16 rows x 32)
        {
            const int r = lane >> 1, half = lane & 1;
            const int ni = nidx[(size_t)tile * 16 + r];
            const float* src =
                cached + (((size_t)(tile >> 16)) * N_PTS + (size_t)ni) * 32 + half * 16;
            #pragma unroll
            for (int j = 0; j < 4; ++j) {
                const float4 q = ((const float4*)src)[j];
                *(float4*)&myT[r * TS + half * 16 + j * 4] = q;
            }
        }
        // ---- GEMM3: feat[16x64] @ Wn[64x64] ----
        v8f c3[4]; c3[0] = vzero8(); c3[1] = vzero8(); c3[2] = vzero8(); c3[3] = vzero8();
        #pragma unroll
        for (int kst = 0; kst < 16; ++kst) {
            const v2f a = *(const v2f*)&myT[col * TS + kst * 4 + 2 * hi];
            #pragma unroll
            for (int ct = 0; ct < 4; ++ct)
                c3[ct] = wmma4(a, bfrag(sWn, kst * 2 + hi, ct * 16 + col, 64), c3[ct]);
        }
        #pragma unroll
        for (int ct = 0; ct < 4; ++ct) {
            const float bb = sBn[ct * 16 + col];
            #pragma unroll
            for (int v = 0; v < 8; ++v) c3[ct][v] += bb;
        }
        if (STAGE == 3) {
            #pragma unroll
            for (int ct = 0; ct < 4; ++ct) {
                float s = 0.f, q = 0.f;
                #pragma unroll
                for (int v = 0; v < 8; ++v) { float x = c3[ct][v]; s += x; q += x * x; }
                ac_s[ct] += s; ac_q[ct] += q;
            }
            continue;
        }
        // ---- STAGE 4: feat3 = lrelu(affine3(y3)), overwrite tile ----
        #pragma unroll
        for (int ct = 0; ct < 4; ++ct) {
            const float sc = sSc3[ct * 16 + col], sh = sSh3[ct * 16 + col];
            #pragma unroll
            for (int v = 0; v < 8; ++v)
                myT[(v + 8 * hi) * TS + ct * 16 + col] = lrelu(c3[ct][v] * sc + sh);
        }
        // logits = feat3 @ Wattn
        v8f l4[4]; l4[0] = vzero8(); l4[1] = vzero8(); l4[2] = vzero8(); l4[3] = vzero8();
        #pragma unroll
        for (int kst = 0; kst < 16; ++kst) {
            const v2f a = *(const v2f*)&myT[col * TS + kst * 4 + 2 * hi];
            #pragma unroll
            for (int ct = 0; ct < 4; ++ct)
                l4[ct] = wmma4(a, bfrag(sWat, kst * 2 + hi, ct * 16 + col, 64), l4[ct]);
        }
        // softmax over the 16 neighbor rows (per column) + weighted pooling
        #pragma unroll
        for (int ct = 0; ct < 4; ++ct) {
            float mx = l4[ct][0];
            #pragma unroll
            for (int v = 1; v < 8; ++v) mx = fmaxf(mx, l4[ct][v]);
            mx = fmaxf(mx, __shfl_xor(mx, 16, 32));
            float e[8]; float sum = 0.f;
            #pragma unroll
            for (int v = 0; v < 8; ++v) { e[v] = __expf(l4[ct][v] - mx); sum += e[v]; }
            sum += __shfl_xor(sum, 16, 32);
            const float inv = 1.f / sum;
            float p = 0.f;
            #pragma unroll
            for (int v = 0; v < 8; ++v) p += e[v] * myT[(v + 8 * hi) * TS + ct * 16 + col];
            p *= inv;
            p += __shfl_xor(p, 16, 32);
            if (hi == 0) pooled[(size_t)tile * 64 + ct * 16 + col] = p;
        }
    }

    if (STAGE < 4) {
        const int CT = (STAGE >= 3) ? 4 : 2;
        for (int ct = 0; ct < CT; ++ct) {
            float s0 = ac_s[ct] + __shfl_xor(ac_s[ct], 16, 32);
            float q0 = ac_q[ct] + __shfl_xor(ac_q[ct], 16, 32);
            if (hi == 0) {
                atomicAdd(&sSum[ct * 16 + col], s0);
                atomicAdd(&sSsq[ct * 16 + col], q0);
            }
        }
        __syncthreads();
        const int C = CT * 16;
        if (t < C) { atomicAdd(&acc[t], sSum[t]); atomicAdd(&acc[64 + t], sSsq[t]); }
    }
}

// ---------------------------------------------------------------------------
// Stats pass for the output heads: Y = X[rows x CIN] @ W[CIN x 64] + b
// ---------------------------------------------------------------------------
template<int CIN>
__global__ __launch_bounds__(256) void gemm_stats_kernel(
    const float* __restrict__ X, const float* __restrict__ W,
    const float* __restrict__ bias, float* __restrict__ acc)
{
    __shared__ float sW[CIN * 64];
    __shared__ float sB[64];
    __shared__ float sT[NW][16 * TS];
    __shared__ float sSum[64], sSsq[64];
    const int t = threadIdx.x;
    for (int i = t; i < CIN * 64; i += 256) {
        const int k = i >> 6, c = i & 63;
        sW[(((k >> 1) * 64) + c) * 2 + (k & 1)] = W[i];
    }
    if (t < 64) { sB[t] = bias[t]; sSum[t] = 0.f; sSsq[t] = 0.f; }
    __syncthreads();
    const int lane = t & 31, wid = t >> 5;
    const int col = lane & 15, hi = lane >> 4;
    float* myT = sT[wid];
    float ac_s[4] = {0, 0, 0, 0}, ac_q[4] = {0, 0, 0, 0};
    constexpr int C4 = CIN / 4;
    for (int tile = blockIdx.x * NW + wid; tile < NTILES_SMALL; tile += gridDim.x * NW) {
        for (int j = lane; j < 16 * C4; j += 32) {
            const int r = j / C4, cc = (j % C4) * 4;
            *(float4*)&myT[r * TS + cc] =
                *(const float4*)&X[(size_t)tile * 16 * CIN + r * CIN + cc];
        }
        v8f c[4]; c[0] = vzero8(); c[1] = vzero8(); c[2] = vzero8(); c[3] = vzero8();
        #pragma unroll
        for (int kst = 0; kst < CIN / 4; ++kst) {
            const v2f a = *(const v2f*)&myT[col * TS + kst * 4 + 2 * hi];
            #pragma unroll
            for (int ct = 0; ct < 4; ++ct)
                c[ct] = wmma4(a, bfrag(sW, kst * 2 + hi, ct * 16 + col, 64), c[ct]);
        }
        #pragma unroll
        for (int ct = 0; ct < 4; ++ct) {
            const float bb = sB[ct * 16 + col];
            float s = 0.f, q = 0.f;
            #pragma unroll
            for (int v = 0; v < 8; ++v) { float x = c[ct][v] + bb; s += x; q += x * x; }
            ac_s[ct] += s; ac_q[ct] += q;
        }
    }
    #pragma unroll
    for (int ct = 0; ct < 4; ++ct) {
        float s0 = ac_s[ct] + __shfl_xor(ac_s[ct], 16, 32);
        float q0 = ac_q[ct] + __shfl_xor(ac_q[ct], 16, 32);
        if (hi == 0) {
            atomicAdd(&sSum[ct * 16 + col], s0);
            atomicAdd(&sSsq[ct * 16 + col], q0);
        }
    }
    __syncthreads();
    if (t < 64) { atomicAdd(&acc[t], sSum[t]); atomicAdd(&acc[64 + t], sSsq[t]); }
}

// ---------------------------------------------------------------------------
// Final: y = lrelu(affine_o(pooled@Wo+bo) + affine_s(cached@Ws+bs))
// ---------------------------------------------------------------------------
__global__ __launch_bounds__(256) void final_kernel(
    const float* __restrict__ pooled, const float* __restrict__ cached,
    const float* __restrict__ Wo, const float* __restrict__ bo,
    const float* __restrict__ Ws, const float* __restrict__ bs,
    const float* __restrict__ ss, float* __restrict__ out)
{
    __shared__ float sWo[64 * 64];
    __shared__ float sWs[32 * 64];
    __shared__ float sBo[64], sBs[64], sSco[64], sSho[64], sScs[64], sShs[64];
    __shared__ float sT[NW][16 * TS];
    const int t = threadIdx.x;
    for (int i = t; i < 64 * 64; i += 256) {
        const int k = i >> 6, c = i & 63;
        sWo[(((k >> 1) * 64) + c) * 2 + (k & 1)] = Wo[i];
    }
    for (int i = t; i < 32 * 64; i += 256) {
        const int k = i >> 6, c = i & 63;
        sWs[(((k >> 1) * 64) + c) * 2 + (k & 1)] = Ws[i];
    }
    if (t < 64) {
        sBo[t] = bo[t]; sBs[t] = bs[t];
        sSco[t] = ss[384 + t]; sSho[t] = ss[448 + t];
        sScs[t] = ss[512 + t]; sShs[t] = ss[576 + t];
    }
    __syncthreads();
    const int lane = t & 31, wid = t >> 5;
    const int col = lane & 15, hi = lane >> 4;
    float* myT = sT[wid];
    for (int tile = blockIdx.x * NW + wid; tile < NTILES_SMALL; tile += gridDim.x * NW) {
        // pooled @ Wo
        for (int j = lane; j < 16 * 16; j += 32) {
            const int r = j >> 4, cc = (j & 15) * 4;
            *(float4*)&myT[r * TS + cc] =
                *(const float4*)&pooled[(size_t)tile * 1024 + r * 64 + cc];
        }
        v8f co[4]; co[0] = vzero8(); co[1] = vzero8(); co[2] = vzero8(); co[3] = vzero8();
        #pragma unroll
        for (int kst = 0; kst < 16; ++kst) {
            const v2f a = *(const v2f*)&myT[col * TS + kst * 4 + 2 * hi];
            #pragma unroll
            for (int ct = 0; ct < 4; ++ct)
                co[ct] = wmma4(a, bfrag(sWo, kst * 2 + hi, ct * 16 + col, 64), co[ct]);
        }
        #pragma unroll
        for (int ct = 0; ct < 4; ++ct) {
            const float bb = sBo[ct * 16 + col], sc = sSco[ct * 16 + col], sh = sSho[ct * 16 + col];
            #pragma unroll
            for (int v = 0; v < 8; ++v) co[ct][v] = (co[ct][v] + bb) * sc + sh;
        }
        // cached @ Ws
        for (int j = lane; j < 16 * 8; j += 32) {
            const int r = j >> 3, cc = (j & 7) * 4;
            *(float4*)&myT[r * TS + cc] =
                *(const float4*)&cached[(size_t)tile * 512 + r * 32 + cc];
        }
        v8f cs[4]; cs[0] = vzero8(); cs[1] = vzero8(); cs[2] = vzero8(); cs[3] = vzero8();
        #pragma unroll
        for (int kst = 0; kst < 8; ++kst) {
            const v2f a = *(const v2f*)&myT[col * TS + kst * 4 + 2 * hi];
            #pragma unroll
            for (int ct = 0; ct < 4; ++ct)
                cs[ct] = wmma4(a, bfrag(sWs, kst * 2 + hi, ct * 16 + col, 64), cs[ct]);
        }
        #pragma unroll
        for (int ct = 0; ct < 4; ++ct) {
            const float bb = sBs[ct * 16 + col], sc = sScs[ct * 16 + col], sh = sShs[ct * 16 + col];
            #pragma unroll
            for (int v = 0; v < 8; ++v) {
                const float y = co[ct][v] + ((cs[ct][v] + bb) * sc + sh);
                out[((size_t)tile * 16 + v + 8 * hi) * 64 + ct * 16 + col] = lrelu(y);
            }
        }
    }
}

// ---------------------------------------------------------------------------
extern "C" void kernel_launch(void* const* d_in, const int* in_sizes, int n_in,
                              void* d_out, int out_size, void* d_ws, size_t ws_size,
                              hipStream_t stream)
{
    const float* cached = (const float*)d_in[1];
    const float* raw    = (const float*)d_in[2];
    const int*   nidx   = (const int*)d_in[3];
    const float* W1  = (const float*)d_in[4];
    const float* b1  = (const float*)d_in[5];
    const float* g1  = (const float*)d_in[6];
    const float* be1 = (const float*)d_in[7];
    const float* W2  = (const float*)d_in[8];
    const float* b2  = (const float*)d_in[9];
    const float* g2  = (const float*)d_in[10];
    const float* be2 = (const float*)d_in[11];
    const float* Wn  = (const float*)d_in[12];
    const float* bn  = (const float*)d_in[13];
    const float* gn  = (const float*)d_in[14];
    const float* ben = (const float*)d_in[15];
    const float* Wat = (const float*)d_in[16];
    const float* Wo  = (const float*)d_in[17];
    const float* bo  = (const float*)d_in[18];
    const float* go  = (const float*)d_in[19];
    const float* beo = (const float*)d_in[20];
    const float* Ws  = (const float*)d_in[21];
    const float* bs  = (const float*)d_in[22];
    const float* gs  = (const float*)d_in[23];
    const float* bes = (const float*)d_in[24];

    float* ws     = (float*)d_ws;
    float* acc    = ws;            // 5 sets x 128 floats (sum @ +0, sumsq @ +64)
    float* ss     = ws + 1024;     // 5 sets x 128 floats (scale @ +0, shift @ +64)
    float* pooled = ws + 2048;     // [131072, 64] f32 = 32 MB
    float* out    = (float*)d_out;

    const float invB = 1.f / (float)NROWS_BIG;
    const float invS = 1.f / (float)NROWS_SMALL;

    zero_ws_kernel<<<4, 256, 0, stream>>>(acc, 1024);

    stage_kernel<1><<<1024, 256, 0, stream>>>(raw, cached, nidx, W1, b1, W2, b2, Wn, bn, Wat,
                                              ss, acc + 0, pooled);
    finalize_stats_kernel<<<1, 64, 0, stream>>>(acc + 0, g1, be1, ss + 0, ss + 64, invB, 32);

    stage_kernel<2><<<1024, 256, 0, stream>>>(raw, cached, nidx, W1, b1, W2, b2, Wn, bn, Wat,
                                              ss, acc + 128, pooled);
    finalize_stats_kernel<<<1, 64, 0, stream>>>(acc + 128, g2, be2, ss + 128, ss + 192, invB, 32);

    stage_kernel<3><<<1024, 256, 0, stream>>>(raw, cached, nidx, W1, b1, W2, b2, Wn, bn, Wat,
                                              ss, acc + 256, pooled);
    finalize_stats_kernel<<<1, 64, 0, stream>>>(acc + 256, gn, ben, ss + 256, ss + 320, invB, 64);

    stage_kernel<4><<<1024, 256, 0, stream>>>(raw, cached, nidx, W1, b1, W2, b2, Wn, bn, Wat,
                                              ss, acc + 256, pooled);

    gemm_stats_kernel<64><<<256, 256, 0, stream>>>(pooled, Wo, bo, acc + 384);
    finalize_stats_kernel<<<1, 64, 0, stream>>>(acc + 384, go, beo, ss + 384, ss + 448, invS, 64);

    gemm_stats_kernel<32><<<256, 256, 0, stream>>>(cached, Ws, bs, acc + 512);
    finalize_stats_kernel<<<1, 64, 0, stream>>>(acc + 512, gs, bes, ss + 512, ss + 576, invS, 64);

    final_kernel<<<256, 256, 0, stream>>>(pooled, cached, Wo, bo, Ws, bs, ss, out);

    (void)in_sizes; (void)n_in; (void)out_size; (void)ws_size;
}